// MPNEncoder_68822555951733
// MI455X (gfx1250) — compile-verified
//
#include <hip/hip_runtime.h>

// MPNEncoder: N=100000 atoms, NB=16, D=H=128, DEPTH=6, relu, residual (dense=False)
// out = relu(concat(x, gather_sum(msg)) @ W_o + b_o)
//
// ws layout: inp[N*H] | msg0[N*H] | msg1[N*H]   (3 * 51.2 MB fp32, L2-resident)

#define N_ATOMS 100000
#define NBNEI   16
#define Hd      128
#define SLD     132          // padded LDS row stride (floats): conflict-free A-frag loads
#define TM      64           // atom rows per workgroup
#define NBLK    ((N_ATOMS + TM - 1) / TM)

typedef float v2f __attribute__((ext_vector_type(2)));
typedef float v8f __attribute__((ext_vector_type(8)));

// ---- WMMA: C[4](16x16 tiles along M) += As(64x128, LDS) @ W[wrow0..wrow0+127][0..127]
// A-frag (16x4 f32): lanes 0-15 -> M=lane, K=4kk+{0,1}; lanes 16-31 -> M=lane-16, K=4kk+{2,3}
// B-frag (4x16 f32): lanes 0-15 -> N=lane, K=4kk+{0,1}; lanes 16-31 -> N=lane-16, K=4kk+{2,3}
__device__ __forceinline__ void wmma_accum_tile(const float* As,
                                                const float* __restrict__ W,
                                                int wrow0, int n0, int ln, int koff,
                                                v8f C[4])
{
    // this wave's 16-wide weight column block: 32 B-fragments (compiler pipelines these)
    v2f bfrag[32];
#pragma unroll
    for (int kk = 0; kk < 32; ++kk) {
        const float* wp = W + (size_t)(wrow0 + 4 * kk + koff) * Hd + n0 + ln;
        bfrag[kk][0] = wp[0];
        bfrag[kk][1] = wp[Hd];
    }
#pragma unroll
    for (int mt = 0; mt < 4; ++mt) {
        const float* ap = As + (mt * 16 + ln) * SLD + koff;
        v8f c = C[mt];
#pragma unroll
        for (int kk = 0; kk < 32; ++kk) {
            v2f a;
            a[0] = ap[4 * kk + 0];
            a[1] = ap[4 * kk + 1];
            c = __builtin_amdgcn_wmma_f32_16x16x4_f32(
                    false, a, false, bfrag[kk], (short)0, c, false, false);
        }
        C[mt] = c;
    }
}

__device__ __forceinline__ void zero_acc(v8f C[4])
{
#pragma unroll
    for (int mt = 0; mt < 4; ++mt)
#pragma unroll
        for (int v = 0; v < 8; ++v)
            C[mt][v] = 0.0f;
}

// ---- stage a dense 64x128 fp32 tile (row-major, row stride Hd) into LDS
__device__ __forceinline__ void stage_dense_tile(float* As, const float* __restrict__ src,
                                                 int r0, int tid)
{
    for (int idx = tid; idx < TM * 32; idx += 256) {
        int i  = idx >> 5;
        int kq = (idx & 31) << 2;
        int row = r0 + i;
        float4 v = make_float4(0.f, 0.f, 0.f, 0.f);
        if (row < N_ATOMS) v = *(const float4*)(src + (size_t)row * Hd + kq);
        *(float4*)(&As[i * SLD + kq]) = v;
    }
}

// ---- gather-sum 16 neighbor message rows per atom into LDS tile.
// float4-wide: 32 lanes x 16B = one full 512B message row per request
// (global_load_b128), 8 atoms per pass, accumulation in VGPRs.
__device__ __forceinline__ void stage_gather_tile(float* As, int* nidx,
                                                  const float* __restrict__ msgIn,
                                                  const int* __restrict__ a2nei,
                                                  int r0, int tid)
{
    for (int idx = tid; idx < TM * NBNEI; idx += 256) {
        int row = r0 + (idx >> 4);
        nidx[idx] = (row < N_ATOMS) ? a2nei[(size_t)row * NBNEI + (idx & 15)] : 0;
    }
    __syncthreads();
    int a0 = tid >> 5;                 // 8 atoms in flight per pass
    int q  = (tid & 31) << 2;          // column quad 0,4,...,124
    for (int a = a0; a < TM; a += 8) {
        float4 s = make_float4(0.f, 0.f, 0.f, 0.f);
        if (r0 + a < N_ATOMS) {
#pragma unroll
            for (int j = 0; j < NBNEI; ++j) {
                const float4 v = *(const float4*)(msgIn +
                        (size_t)nidx[a * NBNEI + j] * Hd + q);
                s.x += v.x; s.y += v.y; s.z += v.z; s.w += v.w;
            }
        }
        *(float4*)(&As[a * SLD + q]) = s;
    }
}

// =============== kernel 1: inp = X @ W_i + b_i ; msg = relu(inp) ===============
__global__ __launch_bounds__(256) void k_init(const float* __restrict__ X,
                                              const float* __restrict__ Wi,
                                              const float* __restrict__ bi,
                                              float* __restrict__ inp,
                                              float* __restrict__ msg)
{
    __shared__ float As[TM * SLD];
    int tid = threadIdx.x;
    int r0  = blockIdx.x * TM;
    stage_dense_tile(As, X, r0, tid);
    __syncthreads();

    int wv = tid >> 5, lane = tid & 31;
    int n0 = wv * 16, ln = lane & 15, koff = (lane < 16) ? 0 : 2;
    v8f C[4];
    zero_acc(C);
    wmma_accum_tile(As, Wi, 0, n0, ln, koff, C);

    float bias = bi[n0 + ln];
#pragma unroll
    for (int mt = 0; mt < 4; ++mt) {
        int rbase = r0 + mt * 16 + ((lane < 16) ? 0 : 8);
#pragma unroll
        for (int v = 0; v < 8; ++v) {
            int row = rbase + v;
            if (row < N_ATOMS) {
                float val = C[mt][v] + bias;
                size_t o = (size_t)row * Hd + n0 + ln;
                inp[o] = val;
                msg[o] = fmaxf(val, 0.0f);
            }
        }
    }
}

// ====== kernel 2..6: msgOut = relu(inp + gather_sum(msgIn) @ W_h + b_h) ======
__global__ __launch_bounds__(256) void k_step(const float* __restrict__ msgIn,
                                              const int* __restrict__ a2nei,
                                              const float* __restrict__ Wh,
                                              const float* __restrict__ bh,
                                              const float* __restrict__ inp,
                                              float* __restrict__ msgOut)
{
    __shared__ float As[TM * SLD];
    __shared__ int   nidx[TM * NBNEI];
    int tid = threadIdx.x;
    int r0  = blockIdx.x * TM;
    stage_gather_tile(As, nidx, msgIn, a2nei, r0, tid);
    __syncthreads();

    int wv = tid >> 5, lane = tid & 31;
    int n0 = wv * 16, ln = lane & 15, koff = (lane < 16) ? 0 : 2;
    v8f C[4];
    zero_acc(C);
    wmma_accum_tile(As, Wh, 0, n0, ln, koff, C);

    float bias = bh[n0 + ln];
#pragma unroll
    for (int mt = 0; mt < 4; ++mt) {
        int rbase = r0 + mt * 16 + ((lane < 16) ? 0 : 8);
#pragma unroll
        for (int v = 0; v < 8; ++v) {
            int row = rbase + v;
            if (row < N_ATOMS) {
                size_t o = (size_t)row * Hd + n0 + ln;
                float val = C[mt][v] + bias + inp[o];
                msgOut[o] = fmaxf(val, 0.0f);
            }
        }
    }
}

// ==== kernel 7: out = relu(concat(X, gather_sum(msg)) @ W_o + b_o), K = 256 ====
__global__ __launch_bounds__(256) void k_final(const float* __restrict__ msgIn,
                                               const float* __restrict__ X,
                                               const int* __restrict__ a2nei,
                                               const float* __restrict__ Wo,
                                               const float* __restrict__ bo,
                                               float* __restrict__ out)
{
    __shared__ float As0[TM * SLD];      // init_messages tile (K rows 0..127 of W_o)
    __shared__ float As1[TM * SLD];      // gathered message tile (K rows 128..255)
    __shared__ int   nidx[TM * NBNEI];
    int tid = threadIdx.x;
    int r0  = blockIdx.x * TM;
    stage_dense_tile(As0, X, r0, tid);
    stage_gather_tile(As1, nidx, msgIn, a2nei, r0, tid);
    __syncthreads();

    int wv = tid >> 5, lane = tid & 31;
    int n0 = wv * 16, ln = lane & 15, koff = (lane < 16) ? 0 : 2;
    v8f C[4];
    zero_acc(C);
    wmma_accum_tile(As0, Wo, 0,   n0, ln, koff, C);
    wmma_accum_tile(As1, Wo, 128, n0, ln, koff, C);

    float bias = bo[n0 + ln];
#pragma unroll
    for (int mt = 0; mt < 4; ++mt) {
        int rbase = r0 + mt * 16 + ((lane < 16) ? 0 : 8);
#pragma unroll
        for (int v = 0; v < 8; ++v) {
            int row = rbase + v;
            if (row < N_ATOMS) {
                float val = C[mt][v] + bias;
                out[(size_t)row * Hd + n0 + ln] = fmaxf(val, 0.0f);
            }
        }
    }
}

extern "C" void kernel_launch(void* const* d_in, const int* in_sizes, int n_in,
                              void* d_out, int out_size, void* d_ws, size_t ws_size,
                              hipStream_t stream)
{
    const float* X     = (const float*)d_in[0];
    // d_in[1] init_attached_features: unused (attach_fea=False)
    const int*   a2nei = (const int*)d_in[2];
    // d_in[3] a2attached: unused
    const float* Wi = (const float*)d_in[4];
    const float* bi = (const float*)d_in[5];
    const float* Wh = (const float*)d_in[6];
    const float* bh = (const float*)d_in[7];
    const float* Wo = (const float*)d_in[8];
    const float* bo = (const float*)d_in[9];

    float* inp = (float*)d_ws;
    float* m0  = inp + (size_t)N_ATOMS * Hd;
    float* m1  = m0  + (size_t)N_ATOMS * Hd;

    dim3 grid(NBLK), block(256);
    k_init<<<grid, block, 0, stream>>>(X, Wi, bi, inp, m0);
    float* cur = m0;
    float* nxt = m1;
    for (int it = 0; it < 5; ++it) {            // DEPTH-1 message passing steps
        k_step<<<grid, block, 0, stream>>>(cur, a2nei, Wh, bh, inp, nxt);
        float* t = cur; cur = nxt; nxt = t;
    }
    k_final<<<grid, block, 0, stream>>>(cur, X, a2nei, Wo, bo, (float*)d_out);
}